// DeconvLayer_55808805044965
// MI455X (gfx1250) — compile-verified
//
#include <hip/hip_runtime.h>
#include <math.h>

typedef __attribute__((ext_vector_type(2))) float v2f;
typedef __attribute__((ext_vector_type(8))) float v8f;

// Problem constants
constexpr int Bn = 4, CIN = 256, COUT = 128, H = 64, W = 64, KK = 9;
constexpr int HW = H * W;               // 4096
constexpr int M1 = Bn * HW;             // 16384  (rows of mdcn GEMM)
constexpr int K1 = CIN * KK;            // 2304   (K of mdcn GEMM)
constexpr int H2 = 128, W2 = 128;
constexpr float EPS = 1e-5f;

// Workspace layout (float offsets)
constexpr size_t OFF_Y  = 0;                                   // dy  [B*9*HW]
constexpr size_t OFF_X  = OFF_Y  + (size_t)Bn * KK * HW;       // dx  [B*9*HW]
constexpr size_t OFF_MK = OFF_X  + (size_t)Bn * KK * HW;       // mask[B*9*HW]
constexpr size_t OFF_WR = OFF_MK + (size_t)Bn * KK * HW;       // wr  [K1][COUT]
constexpr size_t OFF_WD = OFF_WR + (size_t)K1 * COUT;          // wd  [16][128][128]
constexpr size_t OFF_YM = OFF_WD + (size_t)16 * COUT * COUT;   // y   [M1][128]
constexpr size_t OFF_O2 = OFF_YM + (size_t)M1 * COUT;          // out2[4*M1][128]
constexpr size_t OFF_ST = OFF_O2 + (size_t)4 * M1 * COUT;      // stats: 4*128
constexpr size_t OFF_PT = OFF_ST + 512;                        // partials: 2*16384
constexpr size_t OFF_VT = OFF_PT + 32768;                      // valT[K1][M1]

#define WMMA_F32X4(A, B, C) \
  __builtin_amdgcn_wmma_f32_16x16x4_f32(false, (A), false, (B), (short)0, (C), false, false)

// ---------------------------------------------------------------------------
// Stage 1: 3x3 conv producing per-tap offsets (dy,dx) and sigmoid mask.
// om channel layout: dy_k = ch 2k, dx_k = ch 2k+1 (k<9), mask_k = sigmoid(ch 18+k)
// ---------------------------------------------------------------------------
__global__ __launch_bounds__(256) void conv_off_kernel(
    const float* __restrict__ x, const float* __restrict__ w_off,
    const float* __restrict__ b_off, float* __restrict__ ws) {
  int idx = blockIdx.x * blockDim.x + threadIdx.x;        // [B*27*H*W)
  if (idx >= Bn * 27 * HW) return;
  int w = idx & 63, h = (idx >> 6) & 63;
  int t = idx >> 12;
  int ch = t % 27, b = t / 27;

  float acc = b_off[ch];
  const float* xb = x + (size_t)b * CIN * HW;
  const float* wb = w_off + (size_t)ch * CIN * 9;
  for (int c = 0; c < CIN; ++c) {
    const float* xc = xb + (size_t)c * HW;
    const float* wc = wb + c * 9;
#pragma unroll
    for (int ky = 0; ky < 3; ++ky) {
      int ih = h + ky - 1;
      if (ih < 0 || ih >= H) continue;
#pragma unroll
      for (int kx = 0; kx < 3; ++kx) {
        int iw = w + kx - 1;
        if (iw < 0 || iw >= W) continue;
        acc = fmaf(xc[ih * W + iw], wc[ky * 3 + kx], acc);
      }
    }
  }
  if (ch < 18) {
    int k = ch >> 1;
    size_t o = ((size_t)(b * KK + k)) * HW + h * W + w;
    if (ch & 1) ws[OFF_X + o] = acc;
    else        ws[OFF_Y + o] = acc;
  } else {
    int k = ch - 18;
    size_t o = ((size_t)(b * KK + k)) * HW + h * W + w;
    ws[OFF_MK + o] = 1.0f / (1.0f + expf(-acc));
  }
}

// ---------------------------------------------------------------------------
// Stage 2: bilinear-sampled, masked im2col, stored K-major: valT[k*256+c][m]
// ---------------------------------------------------------------------------
__global__ __launch_bounds__(256) void im2col_kernel(
    const float* __restrict__ x, float* __restrict__ ws) {
  int idx = blockIdx.x * blockDim.x + threadIdx.x;        // [B*9*H*W)
  if (idx >= Bn * KK * HW) return;
  int w = idx & 63, h = (idx >> 6) & 63;
  int t = idx >> 12;
  int k = t % KK, b = t / KK;
  int ky = k / 3, kx = k % 3;

  float dy = ws[OFF_Y + idx], dx = ws[OFF_X + idx], mk = ws[OFF_MK + idx];
  float ys = (float)(h - 1 + ky) + dy;
  float xs = (float)(w - 1 + kx) + dx;
  float y0f = floorf(ys), x0f = floorf(xs);
  float ly = ys - y0f, lx = xs - x0f;
  int y0 = (int)y0f, x0 = (int)x0f;
  int y1 = y0 + 1, x1 = x0 + 1;
  float vy0 = (y0 >= 0 && y0 < H) ? 1.f : 0.f;
  float vy1 = (y1 >= 0 && y1 < H) ? 1.f : 0.f;
  float vx0 = (x0 >= 0 && x0 < W) ? 1.f : 0.f;
  float vx1 = (x1 >= 0 && x1 < W) ? 1.f : 0.f;
  float w00 = (1.f - ly) * (1.f - lx) * vy0 * vx0 * mk;
  float w01 = (1.f - ly) * lx        * vy0 * vx1 * mk;
  float w10 = ly * (1.f - lx)        * vy1 * vx0 * mk;
  float w11 = ly * lx                * vy1 * vx1 * mk;
  int yc0 = min(max(y0, 0), H - 1), yc1 = min(max(y1, 0), H - 1);
  int xc0 = min(max(x0, 0), W - 1), xc1 = min(max(x1, 0), W - 1);
  int a00 = yc0 * W + xc0, a01 = yc0 * W + xc1;
  int a10 = yc1 * W + xc0, a11 = yc1 * W + xc1;

  int m = b * HW + h * W + w;
  const float* xb = x + (size_t)b * CIN * HW;
  float* vt = ws + OFF_VT + (size_t)k * CIN * M1 + m;
  for (int c = 0; c < CIN; ++c) {
    const float* xc = xb + (size_t)c * HW;
    float v = w00 * xc[a00] + w01 * xc[a01] + w10 * xc[a10] + w11 * xc[a11];
    vt[(size_t)c * M1] = v;
  }
}

// ---------------------------------------------------------------------------
// Weight repacks (K-major B matrices)
// ---------------------------------------------------------------------------
__global__ __launch_bounds__(256) void repack_wdcn_kernel(
    const float* __restrict__ w_dcn, float* __restrict__ ws) {
  int idx = blockIdx.x * blockDim.x + threadIdx.x;        // [K1*COUT)
  if (idx >= K1 * COUT) return;
  int o = idx & 127;
  int kf = idx >> 7;                 // kf = k*CIN + c
  int k = kf / CIN, c = kf % CIN;
  ws[OFF_WR + idx] = w_dcn[((size_t)o * CIN + c) * KK + k];
}

__global__ __launch_bounds__(256) void repack_wup_kernel(
    const float* __restrict__ w_up, float* __restrict__ ws) {
  int idx = blockIdx.x * blockDim.x + threadIdx.x;        // [16*128*128)
  if (idx >= 16 * COUT * COUT) return;
  int o = idx & 127;
  int r = idx >> 7;
  int i = r & 127;                   // input channel of deconv
  int tp = r >> 7;                   // tap = ty*4+tx
  ws[OFF_WD + idx] = w_up[((size_t)i * COUT + o) * 16 + tp];
}

// ---------------------------------------------------------------------------
// MDCN GEMM: y[m][n] = sum_k valT[k][m]*wr[k][n] + bias[n]
// 8 waves/block as 4(m) x 2(n); each wave computes a 32x32 tile via
// 2x2 f32-WMMA register tiling (4 accumulators). Block tile: 128m x 64n.
// ---------------------------------------------------------------------------
__global__ __launch_bounds__(256) void mdcn_gemm_kernel(
    const float* __restrict__ ws_c, const float* __restrict__ b_dcn,
    float* __restrict__ ws) {
  const float* valT = ws_c + OFF_VT;
  const float* wr   = ws_c + OFF_WR;
  float* y = ws + OFF_YM;

  int wave = threadIdx.x >> 5, lane = threadIdx.x & 31;
  int half = lane >> 4, l16 = lane & 15;
  int wm = wave & 3, wn = wave >> 2;
  int m0 = blockIdx.y * 128 + wm * 32;
  int n0 = blockIdx.x * 64 + wn * 32;

  v8f acc00 = {0.f, 0.f, 0.f, 0.f, 0.f, 0.f, 0.f, 0.f};
  v8f acc01 = acc00, acc10 = acc00, acc11 = acc00;

  const float* aB = valT + (size_t)(half * 2) * M1 + (m0 + l16);
  const float* bB = wr + (half * 2) * COUT + (n0 + l16);
  for (int k0 = 0; k0 < K1; k0 += 4) {
    size_t ka = (size_t)k0 * M1;
    int kb = k0 * COUT;
    v2f a0, a1, b0, b1;
    a0.x = aB[ka];            a0.y = aB[ka + M1];
    a1.x = aB[ka + 16];       a1.y = aB[ka + M1 + 16];
    b0.x = bB[kb];            b0.y = bB[kb + COUT];
    b1.x = bB[kb + 16];       b1.y = bB[kb + COUT + 16];
    // prefetch the streamed A panel ~64 K-rows ahead (global_prefetch_b8)
    __builtin_prefetch(aB + ka + (size_t)64 * M1, 0, 0);
    acc00 = WMMA_F32X4(a0, b0, acc00);
    acc01 = WMMA_F32X4(a0, b1, acc01);
    acc10 = WMMA_F32X4(a1, b0, acc10);
    acc11 = WMMA_F32X4(a1, b1, acc11);
  }
  float bias0 = b_dcn[n0 + l16];
  float bias1 = b_dcn[n0 + 16 + l16];
#pragma unroll
  for (int j = 0; j < 8; ++j) {
    int r0 = m0 + half * 8 + j;
    int r1 = r0 + 16;
    y[(size_t)r0 * COUT + n0 + l16]      = acc00[j] + bias0;
    y[(size_t)r0 * COUT + n0 + 16 + l16] = acc01[j] + bias1;
    y[(size_t)r1 * COUT + n0 + l16]      = acc10[j] + bias0;
    y[(size_t)r1 * COUT + n0 + 16 + l16] = acc11[j] + bias1;
  }
}

// ---------------------------------------------------------------------------
// Deterministic BN stats: per-channel partial sums, then ordered reduce.
// buf is [rows][128] channel-last.
// ---------------------------------------------------------------------------
__global__ __launch_bounds__(128) void stats_partial_kernel(
    const float* __restrict__ buf, int rowsPerBlock, float* __restrict__ part) {
  int c = threadIdx.x;
  size_t r0 = (size_t)blockIdx.x * rowsPerBlock;
  float s = 0.f, q = 0.f;
  for (int r = 0; r < rowsPerBlock; ++r) {
    float v = buf[(r0 + r) * COUT + c];
    s += v;
    q += v * v;
  }
  part[blockIdx.x * COUT + c] = s;
  part[16384 + blockIdx.x * COUT + c] = q;
}

__global__ __launch_bounds__(128) void stats_reduce_kernel(
    const float* __restrict__ part, float* __restrict__ st, int nb) {
  int c = threadIdx.x;
  float s = 0.f, q = 0.f;
  for (int b = 0; b < nb; ++b) {
    s += part[b * COUT + c];
    q += part[16384 + b * COUT + c];
  }
  st[c] = s;
  st[COUT + c] = q;
}

__global__ __launch_bounds__(256) void bn_relu_inplace_kernel(
    float* __restrict__ buf, const float* __restrict__ st,
    const float* __restrict__ g, const float* __restrict__ bt,
    float invN, int n) {
  int idx = blockIdx.x * blockDim.x + threadIdx.x;
  if (idx >= n) return;
  int c = idx & 127;
  float mean = st[c] * invN;
  float var = st[COUT + c] * invN - mean * mean;
  float v = buf[idx];
  v = g[c] * (v - mean) * rsqrtf(var + EPS) + bt[c];
  buf[idx] = fmaxf(v, 0.f);
}

// ---------------------------------------------------------------------------
// Deconv GEMM: per output-parity class p=(py,px), each output pixel sums 4
// taps x 128 input channels. Each wave computes 16m x 32n (2 n-tiles share
// one A fragment, halving the per-lane A gather traffic). out2[(p*M1+m)][o].
// h_in = ho-1+ti+py (w_up index 3-(2ti+py)), same for w.
// ---------------------------------------------------------------------------
__global__ __launch_bounds__(256) void deconv_gemm_kernel(
    const float* __restrict__ ws_c, float* __restrict__ ws) {
  const float* y  = ws_c + OFF_YM;
  const float* wd = ws_c + OFF_WD;
  float* out2 = ws + OFF_O2;

  int wave = threadIdx.x >> 5, lane = threadIdx.x & 31;
  int half = lane >> 4, l16 = lane & 15;
  int p = blockIdx.z, py = p >> 1, px = p & 1;
  int m0 = (blockIdx.y * 8 + wave) * 16;
  int n0 = blockIdx.x * 32;
  int m = m0 + l16;
  int b = m >> 12, ho = (m >> 6) & 63, wo = m & 63;

  v8f acc0 = {0.f, 0.f, 0.f, 0.f, 0.f, 0.f, 0.f, 0.f};
  v8f acc1 = acc0;
#pragma unroll
  for (int ti = 0; ti < 2; ++ti) {
    int hin = ho - 1 + ti + py;
    int wky = 3 - (2 * ti + py);
    bool vh = (hin >= 0) && (hin < H);
    int hc = min(max(hin, 0), H - 1);
#pragma unroll
    for (int tj = 0; tj < 2; ++tj) {
      int win = wo - 1 + tj + px;
      int wkx = 3 - (2 * tj + px);
      bool valid = vh && (win >= 0) && (win < W);
      int wc = min(max(win, 0), W - 1);
      const float* aRow = y + ((size_t)((b * H + hc) * W + wc)) * COUT + half * 2;
      const float* bB = wd + ((size_t)(wky * 4 + wkx) * COUT + half * 2) * COUT
                           + (n0 + l16);
      for (int c0 = 0; c0 < COUT; c0 += 4) {
        v2f a, b0, b1;
        a.x = valid ? aRow[c0] : 0.f;
        a.y = valid ? aRow[c0 + 1] : 0.f;
        b0.x = bB[c0 * COUT];
        b0.y = bB[(c0 + 1) * COUT];
        b1.x = bB[c0 * COUT + 16];
        b1.y = bB[(c0 + 1) * COUT + 16];
        acc0 = WMMA_F32X4(a, b0, acc0);
        acc1 = WMMA_F32X4(a, b1, acc1);
      }
    }
  }
  size_t base = (size_t)p * M1;
#pragma unroll
  for (int j = 0; j < 8; ++j) {
    size_t r = base + m0 + half * 8 + j;
    out2[r * COUT + n0 + l16]      = acc0[j];
    out2[r * COUT + n0 + 16 + l16] = acc1[j];
  }
}

// ---------------------------------------------------------------------------
// Final: BN2 + ReLU, scatter from channel-last parity layout to NCHW output.
// ---------------------------------------------------------------------------
__global__ __launch_bounds__(256) void final_bn_kernel(
    const float* __restrict__ ws_c, const float* __restrict__ g,
    const float* __restrict__ bt, float* __restrict__ out) {
  const float* out2 = ws_c + OFF_O2;
  const float* st = ws_c + OFF_ST + 256;
  int idx = blockIdx.x * blockDim.x + threadIdx.x;        // [4*M1*128)
  if (idx >= 4 * M1 * COUT) return;
  int c = idx & 127;
  int row = idx >> 7;
  int p = row >> 14;
  int mm = row & 16383;
  int b = mm >> 12, ho = (mm >> 6) & 63, wo = mm & 63;
  int h2 = 2 * ho + (p >> 1), w2 = 2 * wo + (p & 1);
  const float invN = 1.0f / (float)(4 * M1);
  float mean = st[c] * invN;
  float var = st[COUT + c] * invN - mean * mean;
  float v = out2[idx];
  v = g[c] * (v - mean) * rsqrtf(var + EPS) + bt[c];
  out[(((size_t)b * COUT + c) * H2 + h2) * W2 + w2] = fmaxf(v, 0.f);
}

// ---------------------------------------------------------------------------
extern "C" void kernel_launch(void* const* d_in, const int* in_sizes, int n_in,
                              void* d_out, int out_size, void* d_ws, size_t ws_size,
                              hipStream_t stream) {
  const float* x      = (const float*)d_in[0];
  const float* w_off  = (const float*)d_in[1];
  const float* b_off  = (const float*)d_in[2];
  const float* w_dcn  = (const float*)d_in[3];
  const float* b_dcn  = (const float*)d_in[4];
  const float* gamma1 = (const float*)d_in[5];
  const float* beta1  = (const float*)d_in[6];
  const float* w_up   = (const float*)d_in[7];
  const float* gamma2 = (const float*)d_in[8];
  const float* beta2  = (const float*)d_in[9];
  float* out = (float*)d_out;
  float* ws = (float*)d_ws;

  // Stage 1: offset conv
  conv_off_kernel<<<(Bn * 27 * HW + 255) / 256, 256, 0, stream>>>(x, w_off, b_off, ws);
  // Weight repacks (independent)
  repack_wdcn_kernel<<<(K1 * COUT + 255) / 256, 256, 0, stream>>>(w_dcn, ws);
  repack_wup_kernel<<<(16 * COUT * COUT + 255) / 256, 256, 0, stream>>>(w_up, ws);
  // Stage 2: bilinear im2col (K-major)
  im2col_kernel<<<(Bn * KK * HW + 255) / 256, 256, 0, stream>>>(x, ws);
  // Stage 3: MDCN GEMM via WMMA  (grid: N/64 x M/128, 8 waves = 4m x 2n)
  mdcn_gemm_kernel<<<dim3(COUT / 64, M1 / 128), 256, 0, stream>>>(ws, b_dcn, ws);
  // BN1 stats (deterministic two-pass) + apply + ReLU
  stats_partial_kernel<<<128, 128, 0, stream>>>(ws + OFF_YM, M1 / 128, ws + OFF_PT);
  stats_reduce_kernel<<<1, 128, 0, stream>>>(ws + OFF_PT, ws + OFF_ST, 128);
  bn_relu_inplace_kernel<<<(M1 * COUT + 255) / 256, 256, 0, stream>>>(
      ws + OFF_YM, ws + OFF_ST, gamma1, beta1, 1.0f / (float)M1, M1 * COUT);
  // Stage 4: transposed conv via parity-class WMMA GEMM (wave = 16m x 32n)
  deconv_gemm_kernel<<<dim3(COUT / 32, M1 / 128, 4), 256, 0, stream>>>(ws, ws);
  // BN2 stats
  stats_partial_kernel<<<128, 128, 0, stream>>>(ws + OFF_O2, (4 * M1) / 128, ws + OFF_PT);
  stats_reduce_kernel<<<1, 128, 0, stream>>>(ws + OFF_PT, ws + OFF_ST + 256, 128);
  // BN2 + ReLU + scatter to NCHW
  final_bn_kernel<<<(4 * M1 * COUT + 255) / 256, 256, 0, stream>>>(ws, gamma2, beta2, out);
}